// UpsampleFrame_17755394801904
// MI455X (gfx1250) — compile-verified
//
#include <hip/hip_runtime.h>
#include <hip/hip_bf16.h>
#include <math.h>

typedef __attribute__((ext_vector_type(2)))  float v2f;
typedef __attribute__((ext_vector_type(8)))  float v8f;

// Problem sizes (fixed by the reference setup)
constexpr int Bn = 2;
constexpr int Nn = 32768;
constexpr int Ss = 8192;                             // power of two (wrap via AND)

constexpr int THREADS         = 256;
constexpr int WAVES           = THREADS / 32;        // 8 waves (wave32)
constexpr int CHUNKS          = 128;                 // query chunks per batch
constexpr int TILES_PER_CHUNK = (Nn / 16) / CHUNKS;  // 16 tiles of 16 queries
constexpr int TILES_PER_WAVE  = TILES_PER_CHUNK / WAVES; // 2

// Flat, branch-free insert of (s, si) into ascending sorted triple.
// Each stage: one v_cmp whose VCC drives flat v_cndmask swaps of both the
// score pair and the index pair (no min/max -> no NaN-canonicalization ops).
__device__ __forceinline__ void ins3(float s, int si,
                                     float& d0, float& d1, float& d2,
                                     int& i0, int& i1, int& i2) {
    // stage 0: winner stays in slot 0, loser bubbles down
    const bool  c0  = s < d0;
    const float l0  = c0 ? d0 : s;
    const int   li0 = c0 ? i0 : si;
    d0 = c0 ? s  : d0;
    i0 = c0 ? si : i0;
    // stage 1
    const bool  c1  = l0 < d1;
    const float l1  = c1 ? d1 : l0;
    const int   li1 = c1 ? i1 : li0;
    d1 = c1 ? l0  : d1;
    i1 = c1 ? li0 : i1;
    // stage 2
    const bool c2 = l1 < d2;
    d2 = c2 ? l1  : d2;
    i2 = c2 ? li1 : i2;
}

__global__ void knn_upsample_kernel(const float* __restrict__ xyz,
                                    const float* __restrict__ sxyz,
                                    const float* __restrict__ sframe,
                                    float* __restrict__ out) {
    extern __shared__ char smem_raw[];
    float4* sp     = reinterpret_cast<float4*>(smem_raw);                     // [Ss] (x,y,z,|p|^2)
    int*    idxbuf = reinterpret_cast<int*>(smem_raw + Ss * sizeof(float4));  // [WAVES*16*3]

    const int tid  = threadIdx.x;
    const int lane = tid & 31;
    const int wave = tid >> 5;
    const int half = (lane >> 4) & 1;   // 0: lanes 0-15, 1: lanes 16-31
    const int col  = lane & 15;

    const int b     = blockIdx.x / CHUNKS;
    const int chunk = blockIdx.x % CHUNKS;

    const float* sx = sxyz   + (size_t)b * 3 * Ss;
    const float* sf = sframe + (size_t)b * 3 * Ss;
    const float* qp = xyz    + (size_t)b * 3 * Nn;
    float*       ob = out    + (size_t)b * 3 * Nn;

    // ---- Stage all sparse points of this batch into LDS: (x, y, z, |p|^2) ----
    for (int s = tid; s < Ss; s += THREADS) {
        float px = sx[s], py = sx[Ss + s], pz = sx[2 * Ss + s];
        sp[s] = make_float4(px, py, pz, px * px + py * py + pz * pz);
    }
    __syncthreads();

    const v8f czero = {0.f, 0.f, 0.f, 0.f, 0.f, 0.f, 0.f, 0.f};

    for (int t = 0; t < TILES_PER_WAVE; ++t) {
        const int tile = chunk * TILES_PER_CHUNK + wave + t * WAVES;
        const int m0   = tile * 16;
        const int m    = m0 + col;

        // Query coords for this lane's row (also reused in the epilogue).
        const float q0 = qp[m];
        const float q1 = qp[Nn + m];
        const float q2 = qp[2 * Nn + m];

        // A (16x4): K0..2 = -2*query coords, K3 = 1.0 (picks up |p|^2 from B).
        // Lanes 0-15 hold K=0,1; lanes 16-31 hold K=2,3.
        const v2f av = { half ? (-2.0f * q2) : (-2.0f * q0),
                         half ? 1.0f         : (-2.0f * q1) };

        // Running top-3 (smallest score) per row; each lane owns 8 rows (r + 8*half).
        float bd0[8], bd1[8], bd2[8];
        int   bi0[8], bi1[8], bi2[8];
        #pragma unroll
        for (int r = 0; r < 8; ++r) {
            bd0[r] = bd1[r] = bd2[r] = 3.0e38f;
            bi0[r] = bi1[r] = bi2[r] = 0;
        }

        // ---- Sweep all sparse tiles. Score = |p|^2 - 2 x.p (order-equivalent to dist).
        //      D = A x B + 0 in one WMMA; next tile prefetched from LDS. ----
        float4 p = sp[col];
        #pragma unroll 2
        for (int s0 = 0; s0 < Ss; s0 += 16) {
            const float4 pn = sp[((s0 + 16) & (Ss - 1)) + col];   // prefetch next tile

            // B (4x16): K0..2 = sparse coords, K3 = |p|^2.
            // Lanes 0-15 hold K=0,1; lanes 16-31 hold K=2,3.
            const v2f bv = { half ? p.z : p.x,
                             half ? p.w : p.y };

            const v8f d = __builtin_amdgcn_wmma_f32_16x16x4_f32(
                              false, av, false, bv, (short)0, czero, false, false);

            const int si = s0 + col;
            #pragma unroll
            for (int r = 0; r < 8; ++r)
                ins3(d[r], si, bd0[r], bd1[r], bd2[r], bi0[r], bi1[r], bi2[r]);

            p = pn;
        }

        // ---- Butterfly merge of per-lane top-3 across the 16 lanes sharing each row ----
        #pragma unroll
        for (int step = 0; step < 4; ++step) {
            const int mask = 1 << step;   // 1,2,4,8: stays within each 16-lane half
            #pragma unroll
            for (int r = 0; r < 8; ++r) {
                const float e0 = __shfl_xor(bd0[r], mask);
                const float e1 = __shfl_xor(bd1[r], mask);
                const float e2 = __shfl_xor(bd2[r], mask);
                const int   j0 = __shfl_xor(bi0[r], mask);
                const int   j1 = __shfl_xor(bi1[r], mask);
                const int   j2 = __shfl_xor(bi2[r], mask);
                ins3(e0, j0, bd0[r], bd1[r], bd2[r], bi0[r], bi1[r], bi2[r]);
                ins3(e1, j1, bd0[r], bd1[r], bd2[r], bi0[r], bi1[r], bi2[r]);
                ins3(e2, j2, bd0[r], bd1[r], bd2[r], bi0[r], bi1[r], bi2[r]);
            }
        }

        // ---- Publish indices (one lane per half), then epilogue on lanes 0-15 ----
        if (col == 0) {
            const int base = (wave * 16 + half * 8) * 3;
            #pragma unroll
            for (int r = 0; r < 8; ++r) {
                idxbuf[base + r * 3 + 0] = bi0[r];
                idxbuf[base + r * 3 + 1] = bi1[r];
                idxbuf[base + r * 3 + 2] = bi2[r];
            }
        }
        __syncthreads();

        if (half == 0) {
            // Lane `col` handles query row m0+col; recompute exact distances like the reference.
            const int base = (wave * 16 + col) * 3;
            float accx = 0.0f, accy = 0.0f, accz = 0.0f, wsum = 0.0f;
            #pragma unroll
            for (int k = 0; k < 3; ++k) {
                const int    id = idxbuf[base + k];
                const float4 pk = sp[id];
                const float dx = pk.x - q0, dy = pk.y - q1, dz = pk.z - q2;
                float dist = sqrtf(dx * dx + dy * dy + dz * dz);
                dist = fmaxf(dist, 1e-10f);
                const float w = 1.0f / dist;
                wsum += w;
                accx += w * sf[id];
                accy += w * sf[Ss + id];
                accz += w * sf[2 * Ss + id];
            }
            const float inv = 1.0f / wsum;
            ob[m]          = accx * inv;
            ob[Nn + m]     = accy * inv;
            ob[2 * Nn + m] = accz * inv;
        }
        __syncthreads();
    }
}

extern "C" void kernel_launch(void* const* d_in, const int* in_sizes, int n_in,
                              void* d_out, int out_size, void* d_ws, size_t ws_size,
                              hipStream_t stream) {
    (void)in_sizes; (void)n_in; (void)out_size; (void)d_ws; (void)ws_size;
    const float* xyz    = (const float*)d_in[0];
    const float* sxyz   = (const float*)d_in[1];
    const float* sframe = (const float*)d_in[2];
    float*       out    = (float*)d_out;

    const size_t shmem = (size_t)Ss * sizeof(float4) + (size_t)WAVES * 16 * 3 * sizeof(int);
    dim3 grid(Bn * CHUNKS);   // 256 blocks, 8 waves each
    knn_upsample_kernel<<<grid, THREADS, shmem, stream>>>(xyz, sxyz, sframe, out);
}